// GCN_81819126989173
// MI455X (gfx1250) — compile-verified
//
#include <hip/hip_runtime.h>
#include <hip/hip_bf16.h>
#include <stddef.h>

typedef __attribute__((ext_vector_type(2))) float v2f;
typedef __attribute__((ext_vector_type(8))) float v8f;

#define N_NODES 50000
#define N_EDGES 800000
#define F1      128
#define F2      8
#define NGRAPH  100

// ---------------- workspace layout (float offsets into d_ws) ----------------
#define WS_DEG_OUT   0
#define WS_DEG_IN    (WS_DEG_OUT  + N_NODES)
#define WS_NORM_SRC  (WS_DEG_IN   + N_NODES)
#define WS_NORM_DST  (WS_NORM_SRC + N_NODES)
#define WS_H         (WS_NORM_DST + N_NODES)        // 50000*128; reused as h1 after scatter1
#define WS_AGG1      (WS_H        + N_NODES * F1)
#define WS_H2        (WS_AGG1     + N_NODES * F1)
#define WS_AGG2      (WS_H2       + N_NODES * F2)
#define WS_GSUM      (WS_AGG2     + N_NODES * F2)
#define WS_GCNT      (WS_GSUM     + NGRAPH * F2)
#define WS_TOTAL     (WS_GCNT     + 128)

// ---------------------------------------------------------------------------
__global__ void k_zero(float* __restrict__ p, int n) {
    int i = blockIdx.x * blockDim.x + threadIdx.x;
    int stride = gridDim.x * blockDim.x;
    for (; i < n; i += stride) p[i] = 0.0f;
}

__global__ void k_degrees(const int* __restrict__ src, const int* __restrict__ dst,
                          float* __restrict__ deg_out, float* __restrict__ deg_in) {
    int e = blockIdx.x * blockDim.x + threadIdx.x;
    if (e < N_EDGES) {
        atomicAdd(&deg_out[src[e]], 1.0f);
        atomicAdd(&deg_in[dst[e]], 1.0f);
    }
}

__global__ void k_norms(const float* __restrict__ deg_out, const float* __restrict__ deg_in,
                        float* __restrict__ norm_src, float* __restrict__ norm_dst) {
    int i = blockIdx.x * blockDim.x + threadIdx.x;
    if (i < N_NODES) {
        float o = deg_out[i], d = deg_in[i];
        norm_src[i] = (o > 0.0f) ? rsqrtf(fmaxf(o, 1.0f)) : 0.0f;
        norm_dst[i] = (d > 0.0f) ? rsqrtf(fmaxf(d, 1.0f)) : 0.0f;
    }
}

// h = x @ W1   (f32, full precision via V_WMMA_F32_16X16X4_F32)
// One 16x16 output tile per wave; 8 waves/block cover the 8 N-tiles of one M-tile.
// 50000 = 3125 * 16 exactly -> no M guard; EXEC stays all-ones for WMMA.
__global__ void __launch_bounds__(256)
k_gemm1_wmma(const float* __restrict__ x, const float* __restrict__ W1,
             float* __restrict__ h) {
    __shared__ float sW[F1 * F1];                       // 64 KB of the 320 KB WGP LDS
    for (int i = threadIdx.x * 4; i < F1 * F1; i += blockDim.x * 4)
        *reinterpret_cast<float4*>(&sW[i]) = *reinterpret_cast<const float4*>(&W1[i]);
    __syncthreads();

    const int wave  = threadIdx.x >> 5;                 // 0..7 -> N tile
    const int lane  = threadIdx.x & 31;
    const int mtile = blockIdx.x;                       // 0..3124
    const int mrow  = mtile * 16 + (lane & 15);         // A-fragment row
    const int n     = wave * 16 + (lane & 15);          // B/C/D column
    const int kh    = (lane >> 4) * 2;                  // K sub-offset (0 or 2)

    const float* __restrict__ xr = x + (size_t)mrow * F1;
    v8f c = {};
    #pragma unroll
    for (int k0 = 0; k0 < F1; k0 += 4) {
        v2f a, b;
        a.x = xr[k0 + kh];
        a.y = xr[k0 + kh + 1];
        b.x = sW[(k0 + kh)     * F1 + n];
        b.y = sW[(k0 + kh + 1) * F1 + n];
        c = __builtin_amdgcn_wmma_f32_16x16x4_f32(false, a, false, b,
                                                  (short)0, c, false, false);
    }
    const int mbase = mtile * 16 + (lane >> 4) * 8;     // C/D row layout
    #pragma unroll
    for (int v = 0; v < 8; ++v)
        h[(size_t)(mbase + v) * F1 + n] = c[v];
}

// layer-1 message scatter: 32 threads per edge, float4 chunks, coalesced gather
__global__ void k_scatter128(const float* __restrict__ h, const int* __restrict__ src,
                             const int* __restrict__ dst, const float* __restrict__ norm_src,
                             float* __restrict__ agg) {
    int tid = blockIdx.x * blockDim.x + threadIdx.x;    // N_EDGES*32 = 25.6M threads
    if (tid >= N_EDGES * 32) return;
    const int e  = tid >> 5;
    const int c4 = (tid & 31) << 2;
    const int s = src[e], d = dst[e];
    const float ns = norm_src[s];
    int pe = e + 2048; if (pe >= N_EDGES) pe = N_EDGES - 1;
    __builtin_prefetch(&src[pe], 0, 0);                 // global_prefetch_b8
    const float4 hv = *reinterpret_cast<const float4*>(h + (size_t)s * F1 + c4);
    float* out = agg + (size_t)d * F1 + c4;
    atomicAdd(out + 0, hv.x * ns);
    atomicAdd(out + 1, hv.y * ns);
    atomicAdd(out + 2, hv.z * ns);
    atomicAdd(out + 3, hv.w * ns);
}

// h1 = relu(agg1 * norm_dst + b1)   (written into the h buffer, now dead)
__global__ void k_update1(const float* __restrict__ agg, const float* __restrict__ norm_dst,
                          const float* __restrict__ b1, float* __restrict__ h1) {
    int i = blockIdx.x * blockDim.x + threadIdx.x;
    if (i < N_NODES * F1) {
        const int node = i >> 7, f = i & 127;
        h1[i] = fmaxf(agg[i] * norm_dst[node] + b1[f], 0.0f);
    }
}

// h2 = h1 @ W2 (128->8); W2 zero-padded to 16 cols in LDS. Tail guard is wave-uniform.
__global__ void __launch_bounds__(256)
k_gemm2_wmma(const float* __restrict__ h1, const float* __restrict__ W2,
             float* __restrict__ h2) {
    __shared__ float sW[F1 * 16];                       // 8 KB
    for (int i = threadIdx.x; i < F1 * 16; i += blockDim.x) {
        const int row = i >> 4, col = i & 15;
        sW[i] = (col < F2) ? W2[row * F2 + col] : 0.0f;
    }
    __syncthreads();

    const int wave  = threadIdx.x >> 5;
    const int lane  = threadIdx.x & 31;
    const int mtile = blockIdx.x * 8 + wave;            // 0..3127 (3125 valid)
    if (mtile >= N_NODES / 16) return;                  // wave-uniform: EXEC all-ones inside

    const int mrow = mtile * 16 + (lane & 15);
    const int n    = lane & 15;
    const int kh   = (lane >> 4) * 2;

    const float* __restrict__ hr = h1 + (size_t)mrow * F1;
    v8f c = {};
    #pragma unroll
    for (int k0 = 0; k0 < F1; k0 += 4) {
        v2f a, b;
        a.x = hr[k0 + kh];
        a.y = hr[k0 + kh + 1];
        b.x = sW[(k0 + kh)     * 16 + n];
        b.y = sW[(k0 + kh + 1) * 16 + n];
        c = __builtin_amdgcn_wmma_f32_16x16x4_f32(false, a, false, b,
                                                  (short)0, c, false, false);
    }
    if (n < F2) {
        const int mbase = mtile * 16 + (lane >> 4) * 8;
        #pragma unroll
        for (int v = 0; v < 8; ++v)
            h2[(size_t)(mbase + v) * F2 + n] = c[v];
    }
}

// layer-2 message scatter: thread per (edge, feature)
__global__ void k_scatter8(const float* __restrict__ h2, const int* __restrict__ src,
                           const int* __restrict__ dst, const float* __restrict__ norm_src,
                           float* __restrict__ agg) {
    int tid = blockIdx.x * blockDim.x + threadIdx.x;    // N_EDGES*8 = 6.4M threads
    if (tid >= N_EDGES * F2) return;
    const int e = tid >> 3, f = tid & 7;
    const int s = src[e], d = dst[e];
    atomicAdd(&agg[(size_t)d * F2 + f], h2[(size_t)s * F2 + f] * norm_src[s]);
}

// out-node update + per-graph pooling accumulation
__global__ void k_update2_pool(const float* __restrict__ agg, const float* __restrict__ norm_dst,
                               const float* __restrict__ b2, const int* __restrict__ graph_ids,
                               float* __restrict__ gsum, float* __restrict__ gcnt) {
    int i = blockIdx.x * blockDim.x + threadIdx.x;      // N_NODES*8
    if (i < N_NODES * F2) {
        const int node = i >> 3, f = i & 7;
        const float v = agg[i] * norm_dst[node] + b2[f];
        const int g = graph_ids[node];
        atomicAdd(&gsum[g * F2 + f], v);
        if (f == 0) atomicAdd(&gcnt[g], 1.0f);
    }
}

__global__ void k_finalize(const float* __restrict__ gsum, const float* __restrict__ gcnt,
                           float* __restrict__ out) {
    int i = blockIdx.x * blockDim.x + threadIdx.x;
    if (i < NGRAPH * F2) {
        const int g = i >> 3;
        out[i] = gsum[i] / fmaxf(gcnt[g], 1.0f);
    }
}

// ---------------------------------------------------------------------------
extern "C" void kernel_launch(void* const* d_in, const int* in_sizes, int n_in,
                              void* d_out, int out_size, void* d_ws, size_t ws_size,
                              hipStream_t stream) {
    const float* x   = (const float*)d_in[0];
    const float* W1  = (const float*)d_in[1];
    const float* b1  = (const float*)d_in[2];
    const float* W2  = (const float*)d_in[3];
    const float* b2  = (const float*)d_in[4];
    const int*   src = (const int*)d_in[5];
    const int*   dst = (const int*)d_in[6];
    const int*   gid = (const int*)d_in[7];
    float*       out = (float*)d_out;

    float* ws       = (float*)d_ws;
    float* deg_out  = ws + WS_DEG_OUT;
    float* deg_in   = ws + WS_DEG_IN;
    float* norm_src = ws + WS_NORM_SRC;
    float* norm_dst = ws + WS_NORM_DST;
    float* h        = ws + WS_H;         // also h1 after update1
    float* agg1     = ws + WS_AGG1;
    float* h2       = ws + WS_H2;
    float* agg2     = ws + WS_AGG2;
    float* gsum     = ws + WS_GSUM;
    float* gcnt     = ws + WS_GCNT;

    const int TB = 256;

    // zero everything we accumulate into (harness poisons ws with 0xAA)
    k_zero<<<2048, TB, 0, stream>>>(ws, WS_TOTAL);

    k_degrees<<<(N_EDGES + TB - 1) / TB, TB, 0, stream>>>(src, dst, deg_out, deg_in);
    k_norms<<<(N_NODES + TB - 1) / TB, TB, 0, stream>>>(deg_out, deg_in, norm_src, norm_dst);

    k_gemm1_wmma<<<N_NODES / 16, TB, 0, stream>>>(x, W1, h);

    k_scatter128<<<(N_EDGES * 32) / TB, TB, 0, stream>>>(h, src, dst, norm_src, agg1);
    k_update1<<<(N_NODES * F1 + TB - 1) / TB, TB, 0, stream>>>(agg1, norm_dst, b1, h);

    k_gemm2_wmma<<<(N_NODES / 16 + 7) / 8, TB, 0, stream>>>(h, W2, h2);

    k_scatter8<<<(N_EDGES * F2 + TB - 1) / TB, TB, 0, stream>>>(h2, src, dst, norm_src, agg2);
    k_update2_pool<<<(N_NODES * F2 + TB - 1) / TB, TB, 0, stream>>>(agg2, norm_dst, b2, gid,
                                                                   gsum, gcnt);
    k_finalize<<<(NGRAPH * F2 + TB - 1) / TB, TB, 0, stream>>>(gsum, gcnt, out);
}